// SGC_53919019434438
// MI455X (gfx1250) — compile-verified
//
#include <hip/hip_runtime.h>
#include <hip/hip_bf16.h>

// ---------------------------------------------------------------------------
// SGC (Simple Graph Conv), K=2 hops, fused linear + log_softmax head.
// N=100000 nodes, E=3.2M edges, F=128 feats, C=40 classes.
//
// Stage 1: deg[n] = 1 + #incoming edges  ->  dis[n] = rsqrt(deg[n])
// Stage 2: per hop:  xo = (dis^2)*xi  (self-loop)  then
//                    xo[col] += dis[row]*dis[col]*xi[row]   (atomic scatter)
// Stage 3: logits = x @ W^T + b via V_WMMA_F32_16X16X4_F32, then log_softmax.
//          W is zero-padded 40->48 rows once so the WMMA inner loop has no
//          divergent loads (no exec-mask save/restore per K-step).
// ---------------------------------------------------------------------------

typedef float v2f __attribute__((ext_vector_type(2)));
typedef float v8f __attribute__((ext_vector_type(8)));

#define FEAT 128
#define CLS 40
#define CLS_PAD 48

// ---- Stage 1 kernels -------------------------------------------------------

__global__ void sgc_fill_ones(float* __restrict__ deg, int n) {
    int i = blockIdx.x * blockDim.x + threadIdx.x;
    if (i < n) deg[i] = 1.0f;                      // self-loop contribution
}

__global__ void sgc_count_deg(float* __restrict__ deg,
                              const int* __restrict__ col, int ne) {
    int i = blockIdx.x * blockDim.x + threadIdx.x;
    if (i < ne) atomicAdd(&deg[col[i]], 1.0f);
}

__global__ void sgc_rsqrt_inplace(float* __restrict__ dis, int n) {
    int i = blockIdx.x * blockDim.x + threadIdx.x;
    if (i < n) dis[i] = __frsqrt_rn(dis[i]);       // deg >= 1 always
}

// ---- Weight padding: wpad[48][128], rows >= 40 are zero --------------------

__global__ void sgc_pad_weight(float* __restrict__ wpad,
                               const float* __restrict__ w) {
    int i = blockIdx.x * blockDim.x + threadIdx.x;   // 48*128 threads
    if (i < CLS_PAD * FEAT) {
        int row = i >> 7;
        wpad[i] = (row < CLS) ? w[i] : 0.0f;
    }
}

// ---- Stage 2 kernels -------------------------------------------------------

// xo[n][:] = dis[n]^2 * xi[n][:]   (initializes every element of xo)
__global__ void sgc_selfloop_init(float* __restrict__ xo,
                                  const float* __restrict__ xi,
                                  const float* __restrict__ dis, int n) {
    long long i = (long long)blockIdx.x * blockDim.x + threadIdx.x;  // n*32 thr
    if (i >= (long long)n * 32) return;
    int node = (int)(i >> 5);
    int lane = (int)(i & 31);
    float d = dis[node];
    float s = d * d;
    const float4 v = *(const float4*)(xi + (size_t)node * FEAT + lane * 4);
    float4 r;
    r.x = s * v.x; r.y = s * v.y; r.z = s * v.z; r.w = s * v.w;
    *(float4*)(xo + (size_t)node * FEAT + lane * 4) = r;
}

// One wave32 per edge; lane l moves features [4l, 4l+4). Coalesced 512B gather
// per wave + global_atomic_add_f32 scatter (L2-resident: 51MB << 192MB L2).
__global__ void sgc_scatter(float* __restrict__ xo,
                            const float* __restrict__ xi,
                            const float* __restrict__ dis,
                            const int* __restrict__ row,
                            const int* __restrict__ col, int ne) {
    const long long total = (long long)ne * 32;
    const long long stride = (long long)gridDim.x * blockDim.x;
    for (long long i = (long long)blockIdx.x * blockDim.x + threadIdx.x;
         i < total; i += stride) {
        int e = (int)(i >> 5);
        int lane = (int)(i & 31);
        int r = row[e];
        int c = col[e];
        float nrm = dis[r] * dis[c];
        const float4 v = *(const float4*)(xi + (size_t)r * FEAT + lane * 4);
        float* dst = xo + (size_t)c * FEAT + lane * 4;
        atomicAdd(dst + 0, nrm * v.x);
        atomicAdd(dst + 1, nrm * v.y);
        atomicAdd(dst + 2, nrm * v.z);
        atomicAdd(dst + 3, nrm * v.w);
    }
}

// ---- Stage 3: WMMA GEMM + log_softmax -------------------------------------
// Block = 128 threads = 4 waves. Wave w owns 16-row tile (blockIdx*4+w).
// Each wave: 3 accumulators (classes 0-15, 16-31, 32-47; wpad rows >=40 zero),
// K=128 in 32 steps of V_WMMA_F32_16X16X4_F32, A reused across the 3 tiles.
// A lane map (16x4 f32): m = lane&15, k = 2*(lane>>4) + {0,1}  -> float2 load.
// B lane map (4x16 f32): n = lane&15, k = 2*(lane>>4) + {0,1}.
// D (16x16 f32, 8 VGPRs): lanes 0-15 rows 0-7, lanes 16-31 rows 8-15.
__global__ __launch_bounds__(128) void sgc_gemm_logsoftmax(
    float* __restrict__ out, const float* __restrict__ x,
    const float* __restrict__ wpad, const float* __restrict__ bias, int n) {
    __shared__ float lds[4][16][CLS_PAD];

    const int lane = threadIdx.x & 31;
    const int wave = threadIdx.x >> 5;
    const int tile = blockIdx.x * 4 + wave;     // 16-row tile index
    const int ntiles = n >> 4;                  // n is a multiple of 16

    if (tile < ntiles) {                        // wave-uniform: EXEC all ones
        const int m0 = tile << 4;
        const int m = lane & 15;
        const int kb = (lane >> 4) << 1;        // 0 or 2
        const float* arow = x + (size_t)(m0 + m) * FEAT + kb;
        const float* brow = wpad + (size_t)m * FEAT + kb;  // +0/+2048/+4096 floats

        v8f acc0 = {}; v8f acc1 = {}; v8f acc2 = {};
        for (int k0 = 0; k0 < FEAT; k0 += 4) {
            v2f a  = *(const v2f*)(arow + k0);
            v2f b0 = *(const v2f*)(brow + k0);
            v2f b1 = *(const v2f*)(brow + k0 + 16 * FEAT);
            v2f b2 = *(const v2f*)(brow + k0 + 32 * FEAT);
            acc0 = __builtin_amdgcn_wmma_f32_16x16x4_f32(
                false, a, false, b0, (short)0, acc0, false, false);
            acc1 = __builtin_amdgcn_wmma_f32_16x16x4_f32(
                false, a, false, b1, (short)0, acc1, false, false);
            acc2 = __builtin_amdgcn_wmma_f32_16x16x4_f32(
                false, a, false, b2, (short)0, acc2, false, false);
        }
        // Spill D to LDS: lanes 0-15 -> rows 0-7, lanes 16-31 -> rows 8-15.
        const int c = lane & 15;
        const int rb = (lane >> 4) << 3;
#pragma unroll
        for (int r = 0; r < 8; ++r) {
            lds[wave][rb + r][c]      = acc0[r];
            lds[wave][rb + r][16 + c] = acc1[r];
            lds[wave][rb + r][32 + c] = acc2[r];
        }
    }
    __syncthreads();

    // 64 rows per block; threads 0-63 each finish one node's log_softmax.
    if (threadIdx.x < 64) {
        const int lrow = threadIdx.x;
        const int grow = blockIdx.x * 64 + lrow;
        if (grow < n) {
            const int wv = lrow >> 4, r = lrow & 15;
            float vals[CLS];
            float mx = -3.402823466e+38f;
#pragma unroll
            for (int cc = 0; cc < CLS; ++cc) {
                float v = lds[wv][r][cc] + bias[cc];
                vals[cc] = v;
                mx = fmaxf(mx, v);
            }
            float s = 0.0f;
#pragma unroll
            for (int cc = 0; cc < CLS; ++cc) s += __expf(vals[cc] - mx);
            const float lse = __logf(s) + mx;
            float* o = out + (size_t)grow * CLS;
#pragma unroll
            for (int cc = 0; cc < CLS; ++cc) o[cc] = vals[cc] - lse;
        }
    }
}

// ---------------------------------------------------------------------------

extern "C" void kernel_launch(void* const* d_in, const int* in_sizes, int n_in,
                              void* d_out, int out_size, void* d_ws,
                              size_t ws_size, hipStream_t stream) {
    const float* feature = (const float*)d_in[0];
    const float* weight  = (const float*)d_in[1];
    const float* bias    = (const float*)d_in[2];
    const int*   ei      = (const int*)d_in[3];
    // d_in[4] = use_feature flag (always 1 in this setup; SGC uses features).

    const int N = in_sizes[0] / FEAT;       // 100000
    const int E = in_sizes[3] / 2;          // 3200000
    const int* rowp = ei;                   // edge_index[0] = sources
    const int* colp = ei + E;               // edge_index[1] = targets
    float* out = (float*)d_out;

    // Workspace: dis[N] | buf0[N*128] | buf1[N*128] | wpad[48*128]  (~103 MB)
    char* ws = (char*)d_ws;
    size_t off = 0;
    float* dis  = (float*)(ws + off);  off += ((size_t)N * 4 + 255) & ~(size_t)255;
    float* buf0 = (float*)(ws + off);  off += (size_t)N * FEAT * 4;
    float* buf1 = (float*)(ws + off);  off += (size_t)N * FEAT * 4;
    float* wpad = (float*)(ws + off);

    const int T = 256;

    // Stage 1: degree -> rsqrt (dis reused for counts, fully rewritten).
    sgc_fill_ones<<<(N + T - 1) / T, T, 0, stream>>>(dis, N);
    sgc_count_deg<<<(E + T - 1) / T, T, 0, stream>>>(dis, colp, E);
    sgc_rsqrt_inplace<<<(N + T - 1) / T, T, 0, stream>>>(dis, N);

    // Zero-padded weight for branch-free WMMA B loads.
    sgc_pad_weight<<<(CLS_PAD * FEAT + T - 1) / T, T, 0, stream>>>(wpad, weight);

    const int selfBlocks = (int)(((long long)N * 32 + T - 1) / T);
    const int scatBlocks = 8192;  // grid-stride over E*32 threads

    // Hop 1: feature -> buf0
    sgc_selfloop_init<<<selfBlocks, T, 0, stream>>>(buf0, feature, dis, N);
    sgc_scatter<<<scatBlocks, T, 0, stream>>>(buf0, feature, dis, rowp, colp, E);

    // Hop 2: buf0 -> buf1
    sgc_selfloop_init<<<selfBlocks, T, 0, stream>>>(buf1, buf0, dis, N);
    sgc_scatter<<<scatBlocks, T, 0, stream>>>(buf1, buf0, dis, rowp, colp, E);

    // Stage 3: WMMA GEMM + log_softmax. 6250 row tiles, 4 waves per block.
    const int ntiles = N / 16;
    const int gBlocks = (ntiles + 3) / 4;
    sgc_gemm_logsoftmax<<<gBlocks, 128, 0, stream>>>(out, buf1, wpad, bias, N);
}